// N3Tree_76596446757433
// MI455X (gfx1250) — compile-verified
//
#include <hip/hip_runtime.h>

// N3-tree query (N=2, DEPTH=6): per-query octree traversal + 16-float feature gather.
// Memory-bound (~304 MB streaming + L2-resident gathers => ~13us floor at 23.3 TB/s).
// CDNA5 path: async global->LDS staging of the hot tree top (levels 0..3, provably
// node<585 at those levels), static LDS/global split per level (pure ds_load on the
// critical chain), 4 queries per thread for 4x memory-level parallelism on the
// dependent lookups, non-temporal hints for the streaming traffic.

#define TREE_DEPTH 6
#define TOP_NODES 585              // levels 0..3 of the complete octree: 1+8+64+512
#define TOP_DWORDS (TOP_NODES * 8) // 4680 dwords per table (18.7 KB); x2 tables = 37.4 KB LDS

typedef float v4f __attribute__((ext_vector_type(4)));
typedef int   v4i __attribute__((ext_vector_type(4)));

// Address-space-qualified v4i pointers for the async-to-LDS builtin
// (param types per hipcc diagnostic: 'int __vector(4) __device__ *').
typedef __attribute__((address_space(1))) v4i* gbl_v4i_p;
typedef __attribute__((address_space(3))) v4i* lds_v4i_p;

#if __has_builtin(__builtin_amdgcn_global_load_async_to_lds_b128)
#define USE_ASYNC_LDS 1
#else
#define USE_ASYNC_LDS 0
#endif

static __device__ __forceinline__ void wait_asynccnt0() {
#if __has_builtin(__builtin_amdgcn_s_wait_asynccnt)
    __builtin_amdgcn_s_wait_asynccnt(0);
#else
    asm volatile("s_wait_asynccnt 0" ::: "memory");
#endif
}

// One traversal level. cptr/dptr are either the LDS mirrors (levels 0..3) or the
// global tables (levels 4..5) — chosen at compile time, so LDS levels lower to ds_load.
static __device__ __forceinline__ void tree_step(float& x, float& y, float& z,
                                                 int& node, int& out_id, bool& active,
                                                 const int* __restrict__ cptr,
                                                 const int* __restrict__ dptr)
{
    x *= 2.0f; y *= 2.0f; z *= 2.0f;
    float fx = fminf(floorf(x), 1.0f);
    float fy = fminf(floorf(y), 1.0f);
    float fz = fminf(floorf(z), 1.0f);
    x -= fx; y -= fy; z -= fz;
    int cell = ((int)fx << 2) | ((int)fy << 1) | (int)fz;
    int e = node * 8 + cell;
    int delta = cptr[e];
    int did   = dptr[e];
    bool term = active && (delta == 0);
    out_id = term ? did : out_id;
    node   = (active && !term) ? (node + delta) : node;
    active = active && !term;
}

__global__ void __launch_bounds__(256)
n3tree_query_kernel(const float* __restrict__ indices,
                    const v4f* __restrict__ features,   // 4 x v4f per 16-float row
                    const int* __restrict__ child,
                    const int* __restrict__ data_idx,
                    const float* __restrict__ invradius,
                    const float* __restrict__ offset,
                    v4f* __restrict__ out,              // 4 x v4f per query
                    int Q)
{
    __shared__ int s_child[TOP_DWORDS];
    __shared__ int s_didx[TOP_DWORDS];

    const int tid = threadIdx.x;

    // ---- Stage top-of-tree tables (levels 0..3) into LDS (ASYNCcnt-tracked DMA) ----
#if USE_ASYNC_LDS
    {
        const int n16 = TOP_DWORDS / 4; // 1170 x 16B per table
        for (int i = tid; i < n16; i += (int)blockDim.x) {
            __builtin_amdgcn_global_load_async_to_lds_b128(
                (gbl_v4i_p)(child + i * 4),
                (lds_v4i_p)(&s_child[i * 4]),
                0, 0);
            __builtin_amdgcn_global_load_async_to_lds_b128(
                (gbl_v4i_p)(data_idx + i * 4),
                (lds_v4i_p)(&s_didx[i * 4]),
                0, 0);
        }
        wait_asynccnt0();
    }
#else
    for (int i = tid; i < TOP_DWORDS; i += (int)blockDim.x) {
        s_child[i] = child[i];
        s_didx[i]  = data_idx[i];
    }
#endif
    __syncthreads();

    // Uniform scale/offset (uniform addresses -> scalar loads)
    const float irx = invradius[0], iry = invradius[1], irz = invradius[2];
    const float ofx = offset[0],    ofy = offset[1],    ofz = offset[2];

    const int stride = (int)(gridDim.x * blockDim.x);
    const int nb = (Q + 3) >> 2;  // batches of 4 queries

    for (int b = (int)(blockIdx.x * blockDim.x) + tid; b < nb; b += stride) {
        const int q0 = b << 2;

        if (q0 + 4 <= Q) {
            // ---- Fast path: 4 queries, 3 aligned 16B NT loads of coords ----
            const v4f* ip = (const v4f*)(indices + (size_t)q0 * 3);
            v4f c0 = __builtin_nontemporal_load(ip + 0); // x0 y0 z0 x1
            v4f c1 = __builtin_nontemporal_load(ip + 1); // y1 z1 x2 y2
            v4f c2 = __builtin_nontemporal_load(ip + 2); // z2 x3 y3 z3

            float xs[4], ys[4], zs[4];
            xs[0] = c0.x; ys[0] = c0.y; zs[0] = c0.z;
            xs[1] = c0.w; ys[1] = c1.x; zs[1] = c1.y;
            xs[2] = c1.z; ys[2] = c1.w; zs[2] = c2.x;
            xs[3] = c2.y; ys[3] = c2.z; zs[3] = c2.w;

            int  nodes[4], ids[4];
            bool act[4];
#pragma unroll
            for (int j = 0; j < 4; ++j) {
                xs[j] = fminf(fmaxf(ofx + xs[j] * irx, 0.0f), 1.0f);
                ys[j] = fminf(fmaxf(ofy + ys[j] * iry, 0.0f), 1.0f);
                zs[j] = fminf(fmaxf(ofz + zs[j] * irz, 0.0f), 1.0f);
                nodes[j] = 0; ids[j] = 0; act[j] = true;
            }

            // 4 interleaved dependency chains; levels 0..3 from LDS, 4..5 from L2.
#pragma unroll
            for (int d = 0; d < TREE_DEPTH; ++d) {
                const int* cp = (d < 4) ? (const int*)s_child : child;
                const int* dp = (d < 4) ? (const int*)s_didx  : data_idx;
#pragma unroll
                for (int j = 0; j < 4; ++j)
                    tree_step(xs[j], ys[j], zs[j], nodes[j], ids[j], act[j], cp, dp);
            }

            // Gather 4 x 64B feature rows, stream out NT.
#pragma unroll
            for (int j = 0; j < 4; ++j) {
                const v4f* frow = features + (size_t)ids[j] * 4;
                v4f f0 = frow[0];
                v4f f1 = frow[1];
                v4f f2 = frow[2];
                v4f f3 = frow[3];
                v4f* orow = out + (size_t)(q0 + j) * 4;
                __builtin_nontemporal_store(f0, orow + 0);
                __builtin_nontemporal_store(f1, orow + 1);
                __builtin_nontemporal_store(f2, orow + 2);
                __builtin_nontemporal_store(f3, orow + 3);
            }
        } else {
            // ---- Tail: per-query scalar path ----
            for (int q = q0; q < Q; ++q) {
                float x = indices[(size_t)q * 3 + 0];
                float y = indices[(size_t)q * 3 + 1];
                float z = indices[(size_t)q * 3 + 2];
                x = fminf(fmaxf(ofx + x * irx, 0.0f), 1.0f);
                y = fminf(fmaxf(ofy + y * iry, 0.0f), 1.0f);
                z = fminf(fmaxf(ofz + z * irz, 0.0f), 1.0f);
                int node = 0, out_id = 0;
                bool active = true;
#pragma unroll
                for (int d = 0; d < TREE_DEPTH; ++d) {
                    const int* cp = (d < 4) ? (const int*)s_child : child;
                    const int* dp = (d < 4) ? (const int*)s_didx  : data_idx;
                    tree_step(x, y, z, node, out_id, active, cp, dp);
                }
                const v4f* frow = features + (size_t)out_id * 4;
                v4f f0 = frow[0], f1 = frow[1], f2 = frow[2], f3 = frow[3];
                v4f* orow = out + (size_t)q * 4;
                __builtin_nontemporal_store(f0, orow + 0);
                __builtin_nontemporal_store(f1, orow + 1);
                __builtin_nontemporal_store(f2, orow + 2);
                __builtin_nontemporal_store(f3, orow + 3);
            }
        }
    }
}

extern "C" void kernel_launch(void* const* d_in, const int* in_sizes, int n_in,
                              void* d_out, int out_size, void* d_ws, size_t ws_size,
                              hipStream_t stream) {
    const float* indices   = (const float*)d_in[0];
    const float* features  = (const float*)d_in[1];
    const int*   child     = (const int*)d_in[2];
    const int*   data_idx  = (const int*)d_in[3];
    const float* invradius = (const float*)d_in[4];
    const float* offset    = (const float*)d_in[5];

    const int Q = in_sizes[0] / 3;

    const int block = 256;                 // 8 wave32 per workgroup
    const int nb = (Q + 3) / 4;            // 4 queries per thread
    int blocks = (nb + block - 1) / block;
    if (blocks > 4096) blocks = 4096;
    if (blocks < 1) blocks = 1;

    n3tree_query_kernel<<<blocks, block, 0, stream>>>(
        indices, (const v4f*)features, child, data_idx,
        invradius, offset, (v4f*)d_out, Q);
}